// MaskedPointGroup_87806311399885
// MI455X (gfx1250) — compile-verified
//
#include <hip/hip_runtime.h>
#include <hip/hip_bf16.h>

#define NPTS 16384
#define NGRP 512
#define GSZ  32
#define NBAT 32
#define CPB  4          // centers per kNN block (4 * 16384 * 4B = 256KB LDS dist)

typedef __attribute__((ext_vector_type(2))) float v2f;
typedef __attribute__((ext_vector_type(8))) float v8f;

// order-preserving float -> uint mapping (ascending)
__device__ __forceinline__ unsigned int fkey(float f) {
    unsigned int b = __float_as_uint(f);
    return b ^ ((((int)b) >> 31) | 0x80000000u);
}

__device__ __forceinline__ unsigned long long shflx64(unsigned long long v, int m) {
    unsigned int lo = (unsigned int)v;
    unsigned int hi = (unsigned int)(v >> 32);
    lo = __shfl_xor(lo, m, 32);
    hi = __shfl_xor(hi, m, 32);
    return ((unsigned long long)hi << 32) | lo;
}

// ---------------------------------------------------------------------------
// Kernel 1: farthest point sampling. One block per batch, 1024 threads,
// 16 points per thread held in VGPRs. Sequential over 512 samples.
// Tie-break: argmax picks lowest index (packed key uses ~idx with max-reduce).
// ---------------------------------------------------------------------------
__global__ void fps_kernel(const float* __restrict__ pc, float* __restrict__ centers) {
    const int b = blockIdx.x;
    const int t = threadIdx.x;              // 0..1023
    const float* base = pc + (size_t)b * NPTS * 6;

    float px[16], py[16], pz[16], md[16];
#pragma unroll
    for (int i = 0; i < 16; ++i) {
        const int p = t * 16 + i;
        const float* s = base + (size_t)p * 6;
        px[i] = s[0]; py[i] = s[1]; pz[i] = s[2];
        md[i] = __builtin_inff();
    }

    __shared__ float bc[3];
    __shared__ int curIdx;
    __shared__ unsigned long long wred[32];

    if (t == 0) curIdx = 0;
    __syncthreads();

    for (int s = 0; s < NGRP; ++s) {
        if (t == 0) {
            const int cur = curIdx;
            const float* cp = base + (size_t)cur * 6;
            const float cx = cp[0], cy = cp[1], cz = cp[2];
            bc[0] = cx; bc[1] = cy; bc[2] = cz;
            float* o = centers + ((size_t)b * NGRP + s) * 3;
            o[0] = cx; o[1] = cy; o[2] = cz;
        }
        __syncthreads();
        if (s == NGRP - 1) break;

        const float cx = bc[0], cy = bc[1], cz = bc[2];
        unsigned long long best = 0ull;
#pragma unroll
        for (int i = 0; i < 16; ++i) {
            const float dx = px[i] - cx;
            const float dy = py[i] - cy;
            const float dz = pz[i] - cz;
            const float d = fmaf(dx, dx, fmaf(dy, dy, dz * dz));
            md[i] = fminf(md[i], d);
            // md >= 0 so raw bits are order-preserving; ~idx => ties pick lowest idx
            const unsigned long long key =
                ((unsigned long long)__float_as_uint(md[i]) << 32) |
                (unsigned int)(~(unsigned int)(t * 16 + i));
            best = (key > best) ? key : best;
        }
        // wave32 max-reduce
#pragma unroll
        for (int o = 16; o > 0; o >>= 1) {
            const unsigned long long v = shflx64(best, o);
            best = (v > best) ? v : best;
        }
        const int lane = t & 31, wv = t >> 5;
        if (lane == 0) wred[wv] = best;
        __syncthreads();
        if (t < 32) {
            unsigned long long v2 = wred[t];
#pragma unroll
            for (int o = 16; o > 0; o >>= 1) {
                const unsigned long long v = shflx64(v2, o);
                v2 = (v > v2) ? v : v2;
            }
            if (t == 0) curIdx = (int)(~(unsigned int)(v2 & 0xFFFFFFFFull));
        }
        __syncthreads();
    }
}

// ---------------------------------------------------------------------------
// Kernel 2: kNN (k=32) + gather. 256 threads, 4 centers per block.
// Distance rank key: |x|^2 - 2 c.x computed with V_WMMA_F32_16X16X4_F32:
//   A row m (m<4 valid): (-2cx, -2cy, -2cz, 1); B col n: (x, y, z, |x|^2).
// A layout (16x4 f32): lanes 0-15 hold {K0,K1}, lanes 16-31 hold {K2,K3}.
// B layout (4x16 f32, mirror):        v.x = row K0 (lanes 0-15) / K2 (16-31),
//                                     v.y = row K1 (lanes 0-15) / K3 (16-31).
// Only lanes 0-15 load each 16-point tile (coalesced 384B); upper half gets
// its (z, |x|^2) rows via ds_bpermute shuffles. Tile loads are software-
// pipelined one tile ahead so the global load overlaps the WMMA + LDS stores.
// Top-32 per center via 32 packed-u64 min extractions (tie -> lowest idx).
// ---------------------------------------------------------------------------
__global__ void knn_kernel(const float* __restrict__ pc,
                           const float* __restrict__ centers,
                           float* __restrict__ nb) {
    extern __shared__ float dist[];         // CPB * NPTS floats (dynamic, 256KB)
    __shared__ unsigned long long wred[8];
    __shared__ int selIdx[CPB * GSZ];
    __shared__ float cc[CPB * 3];

    const int grp = blockIdx.x;             // 0..127
    const int b   = blockIdx.y;             // 0..31
    const int g0  = grp * CPB;
    const int tid = threadIdx.x;            // 0..255
    const int lane = tid & 31, wv = tid >> 5;

    const float* base  = pc + (size_t)b * NPTS * 6;
    const float* cbase = centers + ((size_t)b * NGRP + g0) * 3;

    if (tid < CPB * 3) cc[tid] = cbase[tid];
    __syncthreads();

    // ---- A operand (constant for whole block) ----
    v2f amat;
    {
        const int m = lane & 15;
        float cx = 0.f, cy = 0.f, cz = 0.f, one = 0.f;
        if (m < CPB) {
            cx = cbase[m * 3 + 0];
            cy = cbase[m * 3 + 1];
            cz = cbase[m * 3 + 2];
            one = 1.0f;
        }
        if (lane < 16) { amat.x = -2.f * cx; amat.y = -2.f * cy; }
        else           { amat.x = -2.f * cz; amat.y = one;       }
    }

    // ---- distance tiles: wave wv covers points [wv*2048, wv*2048+2048) ----
    const int wStart = wv * (NPTS / 8);
    const int NT = (NPTS / 8) / 16;         // 128 tiles per wave
    const bool lo = (lane < 16);

    // prologue: load tile 0 (lanes 0-15 only; one coalesced 384B stretch)
    float x = 0.f, y = 0.f, z = 0.f;
    if (lo) {
        const float* s = base + (size_t)(wStart + lane) * 6;
        x = s[0]; y = s[1]; z = s[2];
    }

    for (int tile = 0; tile < NT; ++tile) {
        const int pb = wStart + tile * 16;

        // issue next tile's load before consuming this one (hide latency)
        float nx = 0.f, ny = 0.f, nz = 0.f;
        if (tile + 1 < NT && lo) {
            const float* s = base + (size_t)(pb + 16 + lane) * 6;
            __builtin_prefetch(s + 16 * 6, 0, 1);   // tile t+2, speculative
            nx = s[0]; ny = s[1]; nz = s[2];
        }

        const float sq = fmaf(x, x, fmaf(y, y, z * z));   // valid lanes 0-15
        const float zs  = __shfl(z,  lane & 15, 32);      // ds_bpermute
        const float sqs = __shfl(sq, lane & 15, 32);

        v2f bmat;
        if (lo) { bmat.x = x;  bmat.y = y;   }
        else    { bmat.x = zs; bmat.y = sqs; }

        v8f acc = {};
        acc = __builtin_amdgcn_wmma_f32_16x16x4_f32(
            /*neg_a=*/false, amat, /*neg_b=*/false, bmat,
            /*c_mod=*/(short)0, acc, /*reuse_a=*/false, /*reuse_b=*/false);

        if (lo) {
#pragma unroll
            for (int v = 0; v < CPB; ++v)
                dist[v * NPTS + pb + lane] = acc[v];
        }
        x = nx; y = ny; z = nz;
    }
    __syncthreads();

    // ---- top-32 per center: iterative packed min extraction ----
    for (int c = 0; c < CPB; ++c) {
        float* dc = dist + c * NPTS;
        for (int k = 0; k < GSZ; ++k) {
            unsigned long long m64 = ~0ull;
#pragma unroll 8
            for (int j = 0; j < NPTS / 256; ++j) {
                const int p = tid + j * 256;           // bank-conflict-free
                const unsigned long long key =
                    ((unsigned long long)fkey(dc[p]) << 32) | (unsigned int)p;
                m64 = (key < m64) ? key : m64;
            }
#pragma unroll
            for (int o = 16; o > 0; o >>= 1) {
                const unsigned long long v = shflx64(m64, o);
                m64 = (v < m64) ? v : m64;
            }
            if (lane == 0) wred[wv] = m64;
            __syncthreads();
            if (tid < 32) {
                unsigned long long v2 = (tid < 8) ? wred[tid] : ~0ull;
#pragma unroll
                for (int o = 4; o > 0; o >>= 1) {
                    const unsigned long long v = shflx64(v2, o);
                    v2 = (v < v2) ? v : v2;
                }
                if (tid == 0) {
                    const int idx = (int)(v2 & 0xFFFFFFFFull);
                    selIdx[c * GSZ + k] = idx;
                    dc[idx] = __builtin_inff();        // exclude from next pass
                }
            }
            __syncthreads();
        }
    }

    // ---- gather: neighborhoods[b, g, k, :] with xyz re-centering ----
    if (tid < CPB * GSZ) {
        const int c = tid >> 5, k = tid & 31;
        const int idx = selIdx[c * GSZ + k];
        const int g = g0 + c;
        const float* s = base + (size_t)idx * 6;
        float* o = nb + (((size_t)b * NGRP + g) * GSZ + k) * 6;
        o[0] = s[0] - cc[c * 3 + 0];
        o[1] = s[1] - cc[c * 3 + 1];
        o[2] = s[2] - cc[c * 3 + 2];
        o[3] = s[3];
        o[4] = s[4];
        o[5] = s[5];
    }
}

// ---------------------------------------------------------------------------
extern "C" void kernel_launch(void* const* d_in, const int* in_sizes, int n_in,
                              void* d_out, int out_size, void* d_ws, size_t ws_size,
                              hipStream_t stream) {
    (void)in_sizes; (void)n_in; (void)out_size; (void)d_ws; (void)ws_size;
    const float* pc = (const float*)d_in[0];
    float* out = (float*)d_out;
    float* nb      = out;                                   // [32,512,32,6]
    float* centers = out + (size_t)NBAT * NGRP * GSZ * 6;   // [32,512,3]

    fps_kernel<<<dim3(NBAT), dim3(1024), 0, stream>>>(pc, centers);

    const size_t shmem = (size_t)CPB * NPTS * sizeof(float);   // 256 KB
    static_assert(CPB * NPTS * sizeof(float) == 262144, "lds size");
    hipFuncSetAttribute((const void*)knn_kernel,
                        hipFuncAttributeMaxDynamicSharedMemorySize, (int)shmem);
    knn_kernel<<<dim3(NGRP / CPB, NBAT), dim3(256), shmem, stream>>>(pc, centers, nb);
}